// GraphScaledDotProductAttention_19688130085315
// MI455X (gfx1250) — compile-verified
//
#include <hip/hip_runtime.h>

typedef float v2f __attribute__((ext_vector_type(2)));
typedef float v4f __attribute__((ext_vector_type(4)));
typedef float v8f __attribute__((ext_vector_type(8)));

#define S_LEN 2048
#define DK    64
#define QT    16            // q rows per workgroup
#define LDSP  2052          // padded LDS row stride (floats): conflict-free for WMMA store + b64 A-frag reads
#define NWAVE 8
#define NTHR  (NWAVE * 32)
#define SCALE 0.125f        // 1/sqrt(64)
#define GCOEF (-0.5f)       // -0.5 / (0.5*POS_WIN)^2 with POS_WIN=2.0

__launch_bounds__(NTHR)
__global__ void graph_attn_kernel(const float* __restrict__ qg,
                                  const float* __restrict__ kg,
                                  const float* __restrict__ vg,
                                  const float* __restrict__ bg,
                                  float* __restrict__ og,
                                  float* __restrict__ wg)
{
    __shared__ float sc[QT * LDSP];      // 131328 B: score tile 16 x 2048 (padded)
    __shared__ float red[NTHR];          // partial row reductions
    __shared__ float rowm[QT];           // row max
    __shared__ float rowr[QT];           // 1 / row sum
    __shared__ float obuf[QT * DK];      // output accumulation

    const int tid  = threadIdx.x;
    const int wave = tid >> 5;
    const int lane = tid & 31;
    const int lrow = lane & 15;          // M for A-frag / N for B-frag / N for C
    const int lhi  = lane >> 4;          // half-wave select

    const int nqb  = S_LEN / QT;         // 128 q-blocks per (b,h)
    const int qblk = blockIdx.x % nqb;
    const int bh   = blockIdx.x / nqb;

    const float* q    = qg + (size_t)bh * S_LEN * DK + (size_t)qblk * QT * DK;
    const float* k    = kg + (size_t)bh * S_LEN * DK;
    const float* v    = vg + (size_t)bh * S_LEN * DK;
    const float* bias = bg + (size_t)bh * S_LEN * S_LEN + (size_t)qblk * QT * S_LEN;
    float* outw = wg + (size_t)bh * S_LEN * S_LEN + (size_t)qblk * QT * S_LEN;
    float* outo = og + (size_t)bh * S_LEN * DK + (size_t)qblk * QT * DK;

    // Warm L2 with this workgroup's 128KB bias tile while the WMMA phase runs
    // (global_prefetch_b8; bias is consumed only after phase 1).
    {
        const char* bp = (const char*)bias;
        #pragma unroll
        for (int j = 0; j < 4; ++j)
            __builtin_prefetch(bp + tid * 128 + j * NTHR * 128, 0, 0);
    }

    // ---------- Phase 1: S = (Q*scale) @ K^T  via v_wmma_f32_16x16x4_f32 ----------
    // A-frag layout (16x4 f32): lane L holds A[L%16][2*(L/16) + {0,1}] in {.x,.y}
    v2f aq[16];
    #pragma unroll
    for (int d0 = 0; d0 < 16; ++d0) {
        const float* p = q + lrow * DK + 4 * d0 + 2 * lhi;
        aq[d0].x = p[0] * SCALE;
        aq[d0].y = p[1] * SCALE;
    }

    for (int t = wave; t < S_LEN / 16; t += NWAVE) {
        const int c0 = t * 16;
        // two independent accumulator chains -> 2-way ILP on the matrix pipe
        v8f acc0 = {0.f, 0.f, 0.f, 0.f, 0.f, 0.f, 0.f, 0.f};
        v8f acc1 = {0.f, 0.f, 0.f, 0.f, 0.f, 0.f, 0.f, 0.f};
        // B-frag (4x16 f32): lane L holds B[2*(L/16)+{0,1}][L%16] = K[c0+L%16][depth]
        const float* kp = k + (size_t)(c0 + lrow) * DK + 2 * lhi;
        #pragma unroll
        for (int d0 = 0; d0 < 16; d0 += 2) {
            v2f bk0, bk1;
            bk0.x = kp[4 * d0];
            bk0.y = kp[4 * d0 + 1];
            bk1.x = kp[4 * d0 + 4];
            bk1.y = kp[4 * d0 + 5];
            acc0 = __builtin_amdgcn_wmma_f32_16x16x4_f32(false, aq[d0], false, bk0,
                                                         (short)0, acc0, false, false);
            acc1 = __builtin_amdgcn_wmma_f32_16x16x4_f32(false, aq[d0 + 1], false, bk1,
                                                         (short)0, acc1, false, false);
        }
        v8f acc = acc0 + acc1;
        // C layout: VGPR r -> row (r + 8*lhi), col (c0 + lrow). Conflict-free with LDSP=2052.
        #pragma unroll
        for (int r = 0; r < 8; ++r)
            sc[(r + 8 * lhi) * LDSP + c0 + lrow] = acc[r];
    }
    __syncthreads();

    // ---------- Pass A: fuse Gaussian bias (coalesced NT float4 stream) + row max ----------
    {
        const int row = tid >> 4;        // 16 threads per row
        const int cg  = tid & 15;
        float m = -3.0e38f;
        #pragma unroll 4
        for (int j = 0; j < S_LEN / 64; ++j) {
            const int col = 4 * (cg + 16 * j);
            const v4f b4 =
                __builtin_nontemporal_load((const v4f*)(bias + (size_t)row * S_LEN + col));
            v4f* sp = (v4f*)(sc + row * LDSP + col);
            v4f s4 = *sp;
            s4.x += GCOEF * b4.x * b4.x;
            s4.y += GCOEF * b4.y * b4.y;
            s4.z += GCOEF * b4.z * b4.z;
            s4.w += GCOEF * b4.w * b4.w;
            *sp = s4;
            m = fmaxf(m, fmaxf(fmaxf(s4.x, s4.y), fmaxf(s4.z, s4.w)));
        }
        red[tid] = m;
    }
    __syncthreads();
    if (tid < QT) {
        float m = red[tid * 16];
        #pragma unroll
        for (int i = 1; i < 16; ++i) m = fmaxf(m, red[tid * 16 + i]);
        rowm[tid] = m;
    }
    __syncthreads();

    // ---------- Pass B: p = exp(s - m) in LDS, row sums ----------
    {
        const int row = tid >> 4;
        const int cg  = tid & 15;
        const float m = rowm[row];
        float s = 0.f;
        #pragma unroll 4
        for (int j = 0; j < S_LEN / 64; ++j) {
            const int col = 4 * (cg + 16 * j);
            v4f* sp = (v4f*)(sc + row * LDSP + col);
            v4f s4 = *sp;
            s4.x = __expf(s4.x - m);
            s4.y = __expf(s4.y - m);
            s4.z = __expf(s4.z - m);
            s4.w = __expf(s4.w - m);
            *sp = s4;
            s += s4.x + s4.y + s4.z + s4.w;
        }
        red[tid] = s;
    }
    __syncthreads();
    if (tid < QT) {
        float s = 0.f;
        #pragma unroll
        for (int i = 0; i < 16; ++i) s += red[tid * 16 + i];
        rowr[tid] = 1.0f / s;
    }
    for (int i = tid; i < QT * DK; i += NTHR) obuf[i] = 0.f;
    __syncthreads();

    // ---------- Stream normalized weights out (coalesced NT float4: write-once data) ----------
    {
        const int row = tid >> 4;
        const int cg  = tid & 15;
        const float r = rowr[row];
        #pragma unroll 4
        for (int j = 0; j < S_LEN / 64; ++j) {
            const int col = 4 * (cg + 16 * j);
            v4f s4 = *(const v4f*)(sc + row * LDSP + col);
            s4.x *= r; s4.y *= r; s4.z *= r; s4.w *= r;
            __builtin_nontemporal_store(s4, (v4f*)(outw + (size_t)row * S_LEN + col));
        }
    }

    // ---------- Phase 2: O = P @ V (unnormalized P from LDS; rows scaled by 1/l at end) ----------
    v8f oc[4];
    #pragma unroll
    for (int n = 0; n < 4; ++n)
        #pragma unroll
        for (int r = 0; r < 8; ++r) oc[n][r] = 0.f;

    const int kbeg = wave * (S_LEN / NWAVE);
    for (int kc = kbeg; kc < kbeg + S_LEN / NWAVE; kc += 4) {
        v2f ap;                                   // A = P[16 x 4] from LDS (bank-conflict-free b64)
        const float* lp = sc + lrow * LDSP + kc + 2 * lhi;
        ap.x = lp[0];
        ap.y = lp[1];
        const float* vp = v + (size_t)(kc + 2 * lhi) * DK + lrow;
        #pragma unroll
        for (int n = 0; n < 4; ++n) {             // B = V[4 x 16] per n-tile; 4 independent chains
            v2f bv;
            bv.x = vp[n * 16];
            bv.y = vp[n * 16 + DK];
            oc[n] = __builtin_amdgcn_wmma_f32_16x16x4_f32(false, ap, false, bv,
                                                          (short)0, oc[n], false, false);
        }
    }
    // cross-wave reduction of partial O tiles via LDS float atomics (ds_add_f32)
    #pragma unroll
    for (int n = 0; n < 4; ++n)
        #pragma unroll
        for (int r = 0; r < 8; ++r)
            atomicAdd(&obuf[(r + 8 * lhi) * DK + n * 16 + lrow], oc[n][r]);
    __syncthreads();

    for (int i = tid; i < QT * DK; i += NTHR) {
        const int row = i >> 6;                   // DK == 64
        outo[i] = obuf[i] * rowr[row];
    }
}

extern "C" void kernel_launch(void* const* d_in, const int* in_sizes, int n_in,
                              void* d_out, int out_size, void* d_ws, size_t ws_size,
                              hipStream_t stream)
{
    const float* q = (const float*)d_in[0];
    const float* k = (const float*)d_in[1];
    const float* v = (const float*)d_in[2];
    const float* b = (const float*)d_in[3];

    const int bh = in_sizes[0] / (S_LEN * DK);   // B*H = 16
    float* outo = (float*)d_out;                 // [B,H,S,D] first
    float* outw = outo + (size_t)bh * S_LEN * DK;// then [B,H,S,S]

    dim3 grid(bh * (S_LEN / QT));                // 2048 workgroups
    graph_attn_kernel<<<grid, NTHR, 0, stream>>>(q, k, v, b, outo, outw);
}